// GCN_55147380081105
// MI455X (gfx1250) — compile-verified
//
#include <hip/hip_runtime.h>

#define N_NODES 400
#define BATCH   16
#define F       64
#define EPSF    1e-5f
#define ROWS    (BATCH * N_NODES)              // 6400
#define E_UT    (N_NODES * (N_NODES - 1) / 2)  // 79800

typedef __attribute__((ext_vector_type(2))) float v2f;
typedef __attribute__((ext_vector_type(8))) float v8f;

__device__ __forceinline__ int tri_idx(int s, int d) {
    int i = s < d ? s : d;
    int j = s < d ? d : s;
    return i * N_NODES - (i * (i + 1)) / 2 + (j - i - 1);
}

// ---- degree / normalization precompute --------------------------------------
__global__ void k_deg(const float* __restrict__ data,
                      float* __restrict__ dinv,
                      float* __restrict__ self_norm) {
    int t = blockIdx.x * blockDim.x + threadIdx.x;   // t = b*N + d
    if (t >= ROWS) return;
    int b = t / N_NODES, d = t % N_NODES;
    const float* db = data + (size_t)b * E_UT;
    float deg = 1.0f;
    for (int s = 0; s < N_NODES; ++s) {
        if (s == d) continue;
        deg += (db[tri_idx(s, d)] != 0.0f) ? 1.0f : 0.0f;
    }
    dinv[t]      = rsqrtf(deg);
    self_norm[t] = 1.0f / deg;
}

// ---- build normalized adjacency A[b] (N x N), zero diagonal -----------------
__global__ void k_build_a(const float* __restrict__ data,
                          const float* __restrict__ dinv,
                          float* __restrict__ A) {
    int t = blockIdx.x;                // t = b*N + d  (row of A)
    int b = t / N_NODES, d = t % N_NODES;
    const float* db = data + (size_t)b * E_UT;
    float dd = dinv[t];
    float* row = A + (size_t)t * N_NODES;
    for (int s = threadIdx.x; s < N_NODES; s += blockDim.x) {
        float v = 0.0f;
        if (s != d) {
            float w = (db[tri_idx(s, d)] != 0.0f) ? 1.0f : 0.0f;
            v = w * dinv[b * N_NODES + s] * dd;
        }
        row[s] = v;
    }
}

// ---- one 16x64 output strip via 4x V_WMMA_F32_16X16X4_F32 per k-step --------
// A fragment (16x4): lanes 0-15 hold K=0,1 ; lanes 16-31 hold K=2,3 (row M=lane&15)
// B fragment (4x16): lanes 0-15 hold K rows 0,1 ; lanes 16-31 rows 2,3 (col N=lane&15)
// C/D (16x16): VGPR r -> M=r (lanes 0-15) and M=r+8 (lanes 16-31), N=lane&15
// The A fragment is loaded ONCE per k-step and reused by 4 independent WMMAs.
__device__ __forceinline__ void wmma_strip16x64(const float* Am, int lda,
                                                const float* Bm,       // ldb == 64
                                                float*       Cm,       // ldc == 64
                                                int K, int row0, int lane) {
    const int ldb = F, ldc = F;
    int half = lane >> 4;     // 0 or 1
    int l16  = lane & 15;
    v8f c0 = {}, c1 = {}, c2 = {}, c3 = {};
    const float* ap = Am + (size_t)(row0 + l16) * lda + 2 * half;
    const float* bp = Bm + (size_t)(2 * half) * ldb + l16;
    for (int k = 0; k < K; k += 4) {
        v2f a;
        a.x = ap[k];
        a.y = ap[k + 1];
        const float* bk = bp + (size_t)k * ldb;
        v2f b0, b1, b2, b3;
        b0.x = bk[0];        b0.y = bk[ldb];
        b1.x = bk[16];       b1.y = bk[ldb + 16];
        b2.x = bk[32];       b2.y = bk[ldb + 32];
        b3.x = bk[48];       b3.y = bk[ldb + 48];
        c0 = __builtin_amdgcn_wmma_f32_16x16x4_f32(false, a, false, b0, (short)0, c0, false, false);
        c1 = __builtin_amdgcn_wmma_f32_16x16x4_f32(false, a, false, b1, (short)0, c1, false, false);
        c2 = __builtin_amdgcn_wmma_f32_16x16x4_f32(false, a, false, b2, (short)0, c2, false, false);
        c3 = __builtin_amdgcn_wmma_f32_16x16x4_f32(false, a, false, b3, (short)0, c3, false, false);
    }
    float* cp = Cm + (size_t)(row0 + half * 8) * ldc + l16;
#pragma unroll
    for (int r = 0; r < 8; ++r) {
        cp[(size_t)r * ldc]      = c0[r];
        cp[(size_t)r * ldc + 16] = c1[r];
        cp[(size_t)r * ldc + 32] = c2[r];
        cp[(size_t)r * ldc + 48] = c3[r];
    }
}

// AGG[b] = A[b] (400x400) @ H[b] (400x64).  400 waves (B*25), 4 waves/block.
__global__ void __launch_bounds__(128) k_agg(const float* __restrict__ A,
                                             const float* __restrict__ H,
                                             float* __restrict__ AGG) {
    int wave = (blockIdx.x * blockDim.x + threadIdx.x) >> 5;
    int lane = threadIdx.x & 31;
    int b    = wave / 25;          // 25 row strips per batch
    int rt   = wave % 25;
    wmma_strip16x64(A   + (size_t)b * N_NODES * N_NODES, N_NODES,
                    H   + (size_t)b * N_NODES * F,
                    AGG + (size_t)b * N_NODES * F,
                    N_NODES, rt * 16, lane);
}

// H = X (6400x64) @ W (64x64).  400 waves (400 row strips), 4 waves/block.
__global__ void __launch_bounds__(128) k_linear(const float* __restrict__ X,
                                                const float* __restrict__ W,
                                                float* __restrict__ H) {
    int wave = (blockIdx.x * blockDim.x + threadIdx.x) >> 5;
    int lane = threadIdx.x & 31;
    wmma_strip16x64(X, F, W, H, F, wave * 16, lane);
}

// Layer-0 "x @ W0" with x == ones(B,N,1): broadcast W0 row.
__global__ void k_fill_h0(const float* __restrict__ W0, float* __restrict__ H) {
    int i = blockIdx.x * blockDim.x + threadIdx.x;
    if (i < ROWS * F) H[i] = W0[i & (F - 1)];
}

__global__ void k_zero_stats(float* __restrict__ s) {
    if (threadIdx.x < 128) s[threadIdx.x] = 0.0f;
}

// h = relu(agg + h*self_norm + bias); in-place into AGG; accumulate per-channel
// sum & sumsq (block LDS reduction + global atomics).  16 rows per 256-thr block.
__global__ void k_epilogue(const float* __restrict__ H,
                           float* AGG,                      // read+write (in place)
                           const float* __restrict__ self_norm,
                           const float* __restrict__ bias,
                           float* __restrict__ stats) {
    __shared__ float s_sum[256];
    __shared__ float s_sq[256];
    int ch = threadIdx.x & 63;
    int rg = threadIdx.x >> 6;       // 0..3
    int row0 = blockIdx.x * 16;
    float bv = bias[ch];
    float lsum = 0.0f, lsq = 0.0f;
#pragma unroll
    for (int i = 0; i < 4; ++i) {
        int row = row0 + rg * 4 + i;
        size_t idx = (size_t)row * F + ch;
        float v = AGG[idx] + H[idx] * self_norm[row] + bv;
        v = v > 0.0f ? v : 0.0f;
        AGG[idx] = v;
        lsum += v;
        lsq  += v * v;
    }
    s_sum[threadIdx.x] = lsum;
    s_sq[threadIdx.x]  = lsq;
    __syncthreads();
    if (threadIdx.x < 64) {
        float ts = s_sum[threadIdx.x] + s_sum[threadIdx.x + 64] +
                   s_sum[threadIdx.x + 128] + s_sum[threadIdx.x + 192];
        float tq = s_sq[threadIdx.x] + s_sq[threadIdx.x + 64] +
                   s_sq[threadIdx.x + 128] + s_sq[threadIdx.x + 192];
        atomicAdd(&stats[threadIdx.x], ts);
        atomicAdd(&stats[64 + threadIdx.x], tq);
    }
}

// BatchNorm over (B,N): biased var, per-channel affine.
__global__ void k_bn(const float* __restrict__ HR,
                     const float* __restrict__ stats,
                     const float* __restrict__ gamma,
                     const float* __restrict__ beta,
                     float* __restrict__ X) {
    int i = blockIdx.x * blockDim.x + threadIdx.x;
    if (i >= ROWS * F) return;
    int ch = i & 63;
    const float inv_cnt = 1.0f / (float)ROWS;
    float mean = stats[ch] * inv_cnt;
    float var  = stats[64 + ch] * inv_cnt - mean * mean;
    X[i] = (HR[i] - mean) * rsqrtf(var + EPSF) * gamma[ch] + beta[ch];
}

extern "C" void kernel_launch(void* const* d_in, const int* in_sizes, int n_in,
                              void* d_out, int out_size, void* d_ws, size_t ws_size,
                              hipStream_t stream) {
    (void)in_sizes; (void)n_in; (void)out_size; (void)ws_size;

    const float* data  = (const float*)d_in[0];
    const float* Wl[3] = {(const float*)d_in[1], (const float*)d_in[5], (const float*)d_in[9]};
    const float* bl[3] = {(const float*)d_in[2], (const float*)d_in[6], (const float*)d_in[10]};
    const float* gl[3] = {(const float*)d_in[3], (const float*)d_in[7], (const float*)d_in[11]};
    const float* bt[3] = {(const float*)d_in[4], (const float*)d_in[8], (const float*)d_in[12]};

    float* ws        = (float*)d_ws;
    float* A         = ws;                                  // B*N*N = 2,560,000 f
    float* dinv      = A + (size_t)BATCH * N_NODES * N_NODES;
    float* self_norm = dinv + ROWS;
    float* X         = self_norm + ROWS;                    // ROWS*F
    float* H         = X + (size_t)ROWS * F;                // ROWS*F
    float* AGG       = H + (size_t)ROWS * F;                // ROWS*F (also relu output)
    float* stats     = AGG + (size_t)ROWS * F;              // 128 f

    // Graph normalization precompute
    k_deg<<<(ROWS + 255) / 256, 256, 0, stream>>>(data, dinv, self_norm);
    k_build_a<<<ROWS, 128, 0, stream>>>(data, dinv, A);

    for (int l = 0; l < 3; ++l) {
        if (l == 0)
            k_fill_h0<<<(ROWS * F + 255) / 256, 256, 0, stream>>>(Wl[0], H);
        else
            k_linear<<<100, 128, 0, stream>>>(X, Wl[l], H);

        k_agg<<<100, 128, 0, stream>>>(A, H, AGG);
        k_zero_stats<<<1, 128, 0, stream>>>(stats);
        k_epilogue<<<400, 256, 0, stream>>>(H, AGG, self_norm, bl[l], stats);

        float* outp = (l == 2) ? (float*)d_out : X;
        k_bn<<<(ROWS * F + 255) / 256, 256, 0, stream>>>(AGG, stats, gl[l], bt[l], outp);
    }
}